// FWEnergyGAD_34626026341045
// MI455X (gfx1250) — compile-verified
//
#include <hip/hip_runtime.h>
#include <math.h>

// FWEnergyGAD: elementwise double-well energy + forces + 2x2 Hessian eigendecomp
// + gradient-ascent-dynamics vector. Streaming workload: 52 B/point, ~70 FLOP/pt
// -> HBM-bound (~9.4 us at 23.3 TB/s). No matrix structure => no WMMA; optimize
// for all-b128 coalesced VMEM, NT stores, and raw 1-ulp trans ops (v_sqrt/v_rsq/
// v_rcp) to keep VALU well under the bandwidth roof. wave32, 256-thread blocks.

typedef float f32x2 __attribute__((ext_vector_type(2)));
typedef float f32x4 __attribute__((ext_vector_type(4)));

#define GAD_BETA 0.1f

struct PtOut {
    float e;            // energy
    float fx, fy;       // forces
    float gadx, gady;   // energy_grad (TAU == 1)
    float hxx, hyy;     // hessian diagonal (hxy == BETA constant)
    float l0, l1;       // eigenvalues, ascending
};

__device__ __forceinline__ PtOut compute_pt(float x, float y) {
    PtOut o;
    float u  = x - 0.5f;
    float v  = y - 0.5f;
    float uu = u * u;
    float vv = v * v;
    float au = uu - 1.0f;
    float av = vv - 1.0f;

    o.e = au * au + av * av + GAD_BETA * u * v;

    float gx = 4.0f * u * au + GAD_BETA * v;
    float gy = 4.0f * v * av + GAD_BETA * u;
    o.fx = -gx;
    o.fy = -gy;

    float hxx = 12.0f * uu - 4.0f;
    float hyy = 12.0f * vv - 4.0f;
    o.hxx = hxx;
    o.hyy = hyy;

    // Closed-form symmetric 2x2 eigendecomposition of [[hxx, b], [b, hyy]].
    // Argument d^2 + b^2 >= b^2 = 0.01: never denormal -> raw v_sqrt_f32 is safe.
    float m = 0.5f * (hxx + hyy);
    float d = 0.5f * (hxx - hyy);
    float r = __builtin_amdgcn_sqrtf(d * d + GAD_BETA * GAD_BETA);
    float l0 = m - r;
    float l1 = m + r;
    o.l0 = l0;
    o.l1 = l1;

    // Eigenvector of l0: (b, l0 - hxx); norm^2 >= b^2 = 0.01, well conditioned.
    // Single v_rsq_f32 replaces sqrt + divide.
    float wx = GAD_BETA;
    float wy = l0 - hxx;
    float invn = __builtin_amdgcn_rsqf(wx * wx + wy * wy);
    wx *= invn;
    wy *= invn;

    // gad = -forces + 2 (forces . v0) v0   (sign of v0 irrelevant: projector)
    float dot  = o.fx * wx + o.fy * wy;
    float gadx = -o.fx + 2.0f * dot * wx;
    float gady = -o.fy + 2.0f * dot * wy;

    float det = l0 * l1;
    float mag = __builtin_amdgcn_sqrtf(gadx * gadx + gady * gady);
    if ((det > 0.0f) && (mag < 1.0f)) {
        float s = __builtin_amdgcn_rcpf(fmaxf(mag, 1e-30f));
        gadx *= s;
        gady *= s;
    }
    o.gadx = gadx;
    o.gady = gady;
    return o;
}

// Vectorized path: 4 points per thread. All VMEM is full b128:
//   2x global_load_b128 in, 11x global_store_b128 (NT) out.
// Requires npts % 4 == 0 so every sub-array base (out + k*npts) is 16B aligned.
__global__ __launch_bounds__(256) void gad_kernel_vec4(
        const float* __restrict__ in, float* __restrict__ out, long long npts) {
    const size_t nn    = (size_t)npts;
    const size_t quads = nn >> 2;
    size_t i = (size_t)blockIdx.x * 256u + threadIdx.x;   // quad index
    if (i >= quads) return;

    const f32x4* ip = (const f32x4*)in;
    const f32x4 p0 = ip[2u * i];
    const f32x4 p1 = ip[2u * i + 1u];

    PtOut A = compute_pt(p0.x, p0.y);
    PtOut B = compute_pt(p0.z, p0.w);
    PtOut C = compute_pt(p1.x, p1.y);
    PtOut D = compute_pt(p1.z, p1.w);

    f32x4* oE = (f32x4*)(out);
    f32x4* oF = (f32x4*)(out + nn);
    f32x4* oG = (f32x4*)(out + 3u * nn);
    f32x4* oH = (f32x4*)(out + 5u * nn);
    f32x4* o0 = (f32x4*)(out + 9u * nn);
    f32x4* o1 = (f32x4*)(out + 10u * nn);

    f32x4 e  = {A.e, B.e, C.e, D.e};
    __builtin_nontemporal_store(e, oE + i);

    f32x4 fAB = {A.fx, A.fy, B.fx, B.fy};
    f32x4 fCD = {C.fx, C.fy, D.fx, D.fy};
    __builtin_nontemporal_store(fAB, oF + 2u * i);
    __builtin_nontemporal_store(fCD, oF + 2u * i + 1u);

    f32x4 gAB = {A.gadx, A.gady, B.gadx, B.gady};
    f32x4 gCD = {C.gadx, C.gady, D.gadx, D.gady};
    __builtin_nontemporal_store(gAB, oG + 2u * i);
    __builtin_nontemporal_store(gCD, oG + 2u * i + 1u);

    f32x4 hA = {A.hxx, GAD_BETA, GAD_BETA, A.hyy};
    f32x4 hB = {B.hxx, GAD_BETA, GAD_BETA, B.hyy};
    f32x4 hC = {C.hxx, GAD_BETA, GAD_BETA, C.hyy};
    f32x4 hD = {D.hxx, GAD_BETA, GAD_BETA, D.hyy};
    __builtin_nontemporal_store(hA, oH + 4u * i);
    __builtin_nontemporal_store(hB, oH + 4u * i + 1u);
    __builtin_nontemporal_store(hC, oH + 4u * i + 2u);
    __builtin_nontemporal_store(hD, oH + 4u * i + 3u);

    f32x4 e0 = {A.l0, B.l0, C.l0, D.l0};
    __builtin_nontemporal_store(e0, o0 + i);
    f32x4 e1 = {A.l1, B.l1, C.l1, D.l1};
    __builtin_nontemporal_store(e1, o1 + i);
}

// Scalar fallback for arbitrary n (alignment-safe; not used for B = 4M).
__global__ __launch_bounds__(256) void gad_kernel_scalar(
        const float* __restrict__ in, float* __restrict__ out, long long npts) {
    const size_t nn = (size_t)npts;
    size_t i = (size_t)blockIdx.x * 256u + threadIdx.x;
    if (i >= nn) return;

    float x = in[2u * i];
    float y = in[2u * i + 1u];
    PtOut A = compute_pt(x, y);

    __builtin_nontemporal_store(A.e,      out + i);
    __builtin_nontemporal_store(A.fx,     out + nn + 2u * i);
    __builtin_nontemporal_store(A.fy,     out + nn + 2u * i + 1u);
    __builtin_nontemporal_store(A.gadx,   out + 3u * nn + 2u * i);
    __builtin_nontemporal_store(A.gady,   out + 3u * nn + 2u * i + 1u);
    __builtin_nontemporal_store(A.hxx,    out + 5u * nn + 4u * i);
    __builtin_nontemporal_store(GAD_BETA, out + 5u * nn + 4u * i + 1u);
    __builtin_nontemporal_store(GAD_BETA, out + 5u * nn + 4u * i + 2u);
    __builtin_nontemporal_store(A.hyy,    out + 5u * nn + 4u * i + 3u);
    __builtin_nontemporal_store(A.l0,     out + 9u * nn + i);
    __builtin_nontemporal_store(A.l1,     out + 10u * nn + i);
}

extern "C" void kernel_launch(void* const* d_in, const int* in_sizes, int n_in,
                              void* d_out, int out_size, void* d_ws, size_t ws_size,
                              hipStream_t stream) {
    (void)n_in; (void)out_size; (void)d_ws; (void)ws_size;
    const float* in = (const float*)d_in[0];
    float* out = (float*)d_out;
    long long npts = (long long)in_sizes[0] / 2;   // 4,194,304 points
    if (npts <= 0) return;

    if ((npts & 3LL) == 0) {
        long long quads = npts >> 2;
        int blocks = (int)((quads + 255) / 256);
        gad_kernel_vec4<<<blocks, 256, 0, stream>>>(in, out, npts);
    } else {
        int blocks = (int)((npts + 255) / 256);
        gad_kernel_scalar<<<blocks, 256, 0, stream>>>(in, out, npts);
    }
}